// ExtendableSheafGCNLayer_49838800503522
// MI455X (gfx1250) — compile-verified
//
#include <hip/hip_runtime.h>
#include <stdint.h>

// CDNA5 wave32 WMMA operand types:
//  - V_WMMA_F32_16X16X4_F32: A = 16x4 f32 (2 VGPRs), B = 4x16 f32 (2 VGPRs), C/D = 16x16 f32 (8 VGPRs)
typedef __attribute__((ext_vector_type(2))) float v2f;
typedef __attribute__((ext_vector_type(8))) float v8f;

__global__ void sheaf_zero_out_kernel(float* __restrict__ out, int n) {
    for (int i = blockIdx.x * blockDim.x + threadIdx.x; i < n;
         i += gridDim.x * blockDim.x)
        out[i] = 0.0f;
}

// One wave32 per edge (grid-strided). Per edge:
//   acc1 = A_vu (16x16) x broadcast(x_v)      -> every column of D = h_mid
//   acc2 = rowrep(h_mid) x A_uv (as K x N)    -> every row of D    = h_v
//   atomicAdd(out[u], adj[v,u] * h_v)
// Memory-bound: 2 KB of operator matrices per edge, ~1.02 GB total at 23.3 TB/s
// => ~45 us floor; the 8 f32 WMMAs/edge are far below the matrix-op ceiling.
__launch_bounds__(256, 2)
__global__ void sheaf_edge_wmma_kernel(const float* __restrict__ adj,
                                       const float* __restrict__ emb,
                                       const float* __restrict__ A_uv,
                                       const float* __restrict__ A_vu,
                                       const long long* __restrict__ eidx,
                                       float* __restrict__ out,
                                       int N, int E) {
    const int lane   = threadIdx.x & 31;
    const int laneLo = lane & 15;     // row (A/M) or column (B/N) index
    const int hi     = lane >> 4;     // lane half selects K offset +0 / +2
    const int wave   = blockIdx.x * (blockDim.x >> 5) + (threadIdx.x >> 5);
    const int nwaves = gridDim.x * (blockDim.x >> 5);

    for (int e = wave; e < E; e += nwaves) {   // wave-uniform: EXEC all-ones for WMMA
        const long long u = eidx[e];
        const long long v = eidx[E + e];
        const float* __restrict__ Avu = A_vu + (size_t)e * 256;
        const float* __restrict__ Auv = A_uv + (size_t)e * 256;
        const float* __restrict__ xv  = emb + (size_t)v * 16;

        // Prefetch next edge's 2 KB of operator matrices (global_prefetch_b8).
        if (e + nwaves < E) {
            __builtin_prefetch(A_vu + (size_t)(e + nwaves) * 256 + lane * 8, 0, 3);
            __builtin_prefetch(A_uv + (size_t)(e + nwaves) * 256 + lane * 8, 0, 3);
        }

        // ---- Step 1 operands ------------------------------------------------
        // A (16x4 chunks, k0 = 4c): lane L holds row M=L&15, elems K = 4c + 2*hi + {0,1}
        // B (4x16 chunks): lane half needs x_v[4c + 2*hi + {0,1}] replicated over N
        v2f a[4], b1[4];
#pragma unroll
        for (int c = 0; c < 4; ++c) {
            a[c]  = *(const v2f*)(Avu + laneLo * 16 + 4 * c + 2 * hi); // 8B aligned
            b1[c] = *(const v2f*)(xv + 4 * c + 2 * hi);                // broadcast load
        }

        v8f acc1 = {};
#pragma unroll
        for (int c = 0; c < 4; ++c)
            acc1 = __builtin_amdgcn_wmma_f32_16x16x4_f32(
                false, a[c], false, b1[c], (short)0, acc1, false, false);
        // acc1 VGPR j: lanes 0-15 -> h_mid[j], lanes 16-31 -> h_mid[j+8] (all N equal)

        // Broadcast h_mid[y] wave-wide via v_readlane_b32 (compile-time source
        // lanes 0 / 16). Bitcast through i32: the builtin is (i32,i32)->i32 and
        // a raw float argument would be VALUE-converted (v_cvt_*), not bitcast.
        float hm[16];
#pragma unroll
        for (int y = 0; y < 16; ++y)
            hm[y] = __int_as_float(
                __builtin_amdgcn_readlane(__float_as_int(acc1[y & 7]),
                                          (y >> 3) << 4));

        // ---- Step 2 operands ------------------------------------------------
        // A2[m][y] = h_mid[y]  (row-replicated): VGPRs carry K = 4c + 2*hi + {0,1}
        // B2[y][x] = A_uv[y][x]: K rows straight from row-major memory, N = laneLo
        v2f a2[4], b2[4];
#pragma unroll
        for (int c = 0; c < 4; ++c) {
            a2[c].x = hi ? hm[4 * c + 2] : hm[4 * c + 0];
            a2[c].y = hi ? hm[4 * c + 3] : hm[4 * c + 1];
            const int row = 4 * c + 2 * hi;
            b2[c].x = Auv[row * 16 + laneLo];         // lanes 0-15 / 16-31 read
            b2[c].y = Auv[(row + 1) * 16 + laneLo];   // two contiguous 64B rows
        }

        v8f acc2 = {};
#pragma unroll
        for (int c = 0; c < 4; ++c)
            acc2 = __builtin_amdgcn_wmma_f32_16x16x4_f32(
                false, a2[c], false, b2[c], (short)0, acc2, false, false);
        // Every row of acc2 equals h_v -> lane L holds h_v[L&15] in acc2[0].

        const float w  = adj[(size_t)v * N + (size_t)u]; // adj_matrix[v, u]
        const float hv = acc2[0];

        if (lane < 16)  // lanes 16-31 are duplicates
            atomicAdd(out + (size_t)u * 16 + laneLo, w * hv); // global_atomic_add_f32
    }
}

extern "C" void kernel_launch(void* const* d_in, const int* in_sizes, int n_in,
                              void* d_out, int out_size, void* d_ws, size_t ws_size,
                              hipStream_t stream) {
    const float*     adj  = (const float*)d_in[0];      // [N, N]
    const float*     emb  = (const float*)d_in[1];      // [N, 16]
    const float*     A_uv = (const float*)d_in[2];      // [E, 16, 16]
    const float*     A_vu = (const float*)d_in[3];      // [E, 16, 16]
    const long long* eidx = (const long long*)d_in[4];  // [2, E] int64
    float* out = (float*)d_out;                         // [N, 16]

    const int N = in_sizes[1] / 16;
    const int E = in_sizes[2] / 256;

    sheaf_zero_out_kernel<<<512, 256, 0, stream>>>(out, out_size);

    // 8 waves / block, one edge per wave per iteration. ~8K waves in flight
    // is plenty to saturate 23.3 TB/s with 2 KB streamed per edge.
    sheaf_edge_wmma_kernel<<<1024, 256, 0, stream>>>(adj, emb, A_uv, A_vu,
                                                     eidx, out, N, E);
}